// TransformerEncoderLayer_11957188952535
// MI455X (gfx1250) — compile-verified
//
#include <hip/hip_runtime.h>
#include <math.h>

// ---------------------------------------------------------------------------
// TransformerEncoderLayer forward, fp32, MI455X (gfx1250, wave32).
// All GEMMs use V_WMMA_F32_16X16X4_F32 (fp32 matrix path => exact dtype
// fidelity with the fp32 reference). Attention is flash-style (no NxN
// materialization): online softmax with shfl_xor row reductions matching the
// WMMA C-layout, and an LDS 16x16 transpose to feed P back as an A-fragment.
// ---------------------------------------------------------------------------

#define LAYERS 2
#define NTOK   4096
#define DM     128
#define HEADS  8
#define DQKV   64     // d_qk == d_v
#define DHID   512
#define QKVC   512    // DQKV * HEADS
#define INV_T  0.08838834764831845f   // 1/sqrt(128)  (T = D**0.5 quirk)
#define LN_EPS 1e-5f

typedef __attribute__((ext_vector_type(2))) float v2f;
typedef __attribute__((ext_vector_type(8))) float v8f;

static __device__ __forceinline__ v8f vzero8() {
  v8f v = {0.f, 0.f, 0.f, 0.f, 0.f, 0.f, 0.f, 0.f};
  return v;
}

// D = A(16x4) * B(4x16) + C, fp32.
static __device__ __forceinline__ v8f wmma4(v2f a, v2f b, v8f c) {
  return __builtin_amdgcn_wmma_f32_16x16x4_f32(
      /*neg_a=*/false, a, /*neg_b=*/false, b,
      /*c_mod=*/(short)0, c, /*reuse_a=*/false, /*reuse_b=*/false);
}

// A fragment: 16x4 tile of row-major M[rows][ld] at (m0,k0).
// ISA layout: lane<16: M=lane, K=k0+{0,1}; lane>=16: M=lane-16, K=k0+{2,3}.
static __device__ __forceinline__ v2f load_a(const float* p, int ld, int m0,
                                             int k0, int lane) {
  int row = m0 + (lane & 15);
  int col = k0 + ((lane >> 4) << 1);
  const float* q = p + (size_t)row * ld + col;
  v2f r = {q[0], q[1]};
  return r;
}

// B fragment: 4x16 tile of row-major W[K][ld] at (k0,n0).
// v0 = row (k0 + 2*half), v1 = row (k0 + 2*half + 1), col = n0 + (lane&15).
static __device__ __forceinline__ v2f load_b(const float* p, int ld, int k0,
                                             int n0, int lane) {
  int kk = k0 + ((lane >> 4) << 1);
  int col = n0 + (lane & 15);
  v2f r = {p[(size_t)kk * ld + col], p[(size_t)(kk + 1) * ld + col]};
  return r;
}

// B fragment of W^T where W is row-major [n][ld]: B[k][n] = W[n][k].
// Per lane this is a contiguous pair -> single b64 load.
static __device__ __forceinline__ v2f load_bt(const float* p, int ld, int k0,
                                              int n0, int lane) {
  int kk = k0 + ((lane >> 4) << 1);
  int col = n0 + (lane & 15);
  const float* q = p + (size_t)col * ld + kk;
  v2f r = {q[0], q[1]};
  return r;
}

// Reductions across each 16-lane half (rows of the C layout span 16 lanes).
static __device__ __forceinline__ float half_max(float v) {
  v = fmaxf(v, __shfl_xor(v, 1));
  v = fmaxf(v, __shfl_xor(v, 2));
  v = fmaxf(v, __shfl_xor(v, 4));
  v = fmaxf(v, __shfl_xor(v, 8));
  return v;
}
static __device__ __forceinline__ float half_sum(float v) {
  v += __shfl_xor(v, 1);
  v += __shfl_xor(v, 2);
  v += __shfl_xor(v, 4);
  v += __shfl_xor(v, 8);
  return v;
}

// ---------------------------------------------------------------------------
// Kernel 1: Q/K/V projection [4096,128]x[128,512], de-interleaved per head.
// grid = (512/64, 4096/16, 3), block = 32.
// Output layout: qkvbuf[which][h][n][dqkv], since ref stores head as the
// MINOR axis of the 512 columns (col = k*H + h).
// ---------------------------------------------------------------------------
__global__ void qkv_kernel(const float* __restrict__ x,
                           const float* __restrict__ Wq,
                           const float* __restrict__ Wk,
                           const float* __restrict__ Wv,
                           float* __restrict__ qkvbuf) {
  const int lane = threadIdx.x;
  const int n0 = blockIdx.x * 64;
  const int m0 = blockIdx.y * 16;
  const int which = blockIdx.z;
  const float* W = (which == 0) ? Wq : ((which == 1) ? Wk : Wv);
  float* out = qkvbuf + (size_t)which * HEADS * NTOK * DQKV;

  v8f acc[4];
#pragma unroll
  for (int c = 0; c < 4; ++c) acc[c] = vzero8();

  for (int k0 = 0; k0 < DM; k0 += 4) {
    v2f a = load_a(x, DM, m0, k0, lane);
#pragma unroll
    for (int c = 0; c < 4; ++c)
      acc[c] = wmma4(a, load_b(W, QKVC, k0, n0 + c * 16, lane), acc[c]);
  }

  const int rowoff = (lane >> 4) << 3;
  const int nlo = lane & 15;
#pragma unroll
  for (int c = 0; c < 4; ++c) {
#pragma unroll
    for (int r = 0; r < 8; ++r) {
      int row = m0 + r + rowoff;
      int col = n0 + c * 16 + nlo;
      int h = col & (HEADS - 1);
      int kq = col >> 3;
      out[((size_t)h * NTOK + row) * DQKV + kq] = acc[c][r];
    }
  }
}

// ---------------------------------------------------------------------------
// Kernel 2: flash attention per head. grid = (4096/16, 8), block = 32.
// One wave owns 16 query rows; streams keys 16 at a time with online softmax.
// ---------------------------------------------------------------------------
__global__ void attn_kernel(const float* __restrict__ qkvbuf,
                            float* __restrict__ Obuf) {
  const int lane = threadIdx.x;
  const int q0 = blockIdx.x * 16;
  const int h = blockIdx.y;
  const float* Q = qkvbuf + ((size_t)0 * HEADS + h) * NTOK * DQKV;
  const float* K = qkvbuf + ((size_t)1 * HEADS + h) * NTOK * DQKV;
  const float* V = qkvbuf + ((size_t)2 * HEADS + h) * NTOK * DQKV;

  __shared__ float pt[16 * 18];  // stride 18: bank-conflict-free, 8B aligned

  // Preload all Q A-fragments for this 16-row block (K dim = 64 -> 16 steps).
  v2f qf[16];
#pragma unroll
  for (int kk = 0; kk < 16; ++kk) qf[kk] = load_a(Q, DQKV, q0, kk * 4, lane);

  float m[8], lsum[8];
  v8f oacc[4];
#pragma unroll
  for (int r = 0; r < 8; ++r) { m[r] = -3.0e38f; lsum[r] = 0.f; }
#pragma unroll
  for (int c = 0; c < 4; ++c) oacc[c] = vzero8();

  const int hi = lane >> 4;
  const int lo = lane & 15;

  for (int j0 = 0; j0 < NTOK; j0 += 16) {
    // S tile = Q(16x64) * K_chunk^T(64x16)
    v8f s = vzero8();
#pragma unroll
    for (int kk = 0; kk < 16; ++kk)
      s = wmma4(qf[kk], load_bt(K, DQKV, kk * 4, j0, lane), s);

    float p[8], fsc[8];
#pragma unroll
    for (int r = 0; r < 8; ++r) {
      float sv = s[r] * INV_T;
      float mn = fmaxf(m[r], half_max(sv));
      fsc[r] = __expf(m[r] - mn);
      p[r] = __expf(sv - mn);
      lsum[r] = lsum[r] * fsc[r] + half_sum(p[r]);
      m[r] = mn;
    }
#pragma unroll
    for (int c = 0; c < 4; ++c)
#pragma unroll
      for (int r = 0; r < 8; ++r) oacc[c][r] *= fsc[r];

    // C-layout -> A-layout transpose of P through LDS.
#pragma unroll
    for (int r = 0; r < 8; ++r) pt[(r + hi * 8) * 18 + lo] = p[r];
    __syncthreads();  // single-wave WG: barrier is NOP, forces s_wait_dscnt

    // O += P(16x16) * V_chunk(16x64)
#pragma unroll
    for (int kk = 0; kk < 4; ++kk) {
      int pcol = kk * 4 + hi * 2;
      v2f pa = {pt[lo * 18 + pcol], pt[lo * 18 + pcol + 1]};
#pragma unroll
      for (int c = 0; c < 4; ++c)
        oacc[c] =
            wmma4(pa, load_b(V + (size_t)j0 * DQKV, DQKV, kk * 4, c * 16, lane),
                  oacc[c]);
    }
    __syncthreads();  // protect pt before next chunk's writes
  }

  float linv[8];
#pragma unroll
  for (int r = 0; r < 8; ++r) linv[r] = 1.f / lsum[r];

  // Store in the reference's flattened layout: o_flat[row, dv*H + h].
  const int rowoff = hi << 3;
#pragma unroll
  for (int c = 0; c < 4; ++c) {
#pragma unroll
    for (int r = 0; r < 8; ++r) {
      int row = q0 + r + rowoff;
      int dv = c * 16 + lo;
      Obuf[(size_t)row * QKVC + dv * HEADS + h] = oacc[c][r] * linv[r];
    }
  }
}

// ---------------------------------------------------------------------------
// Kernel 3/5: [N,K]x[K,128] GEMM + optional bias + residual + LayerNorm.
// grid = (4096/16), block = 32. Wave owns full 16x128 rows -> LN row
// reductions are 8 in-register adds + one 16-lane shfl reduction.
// ---------------------------------------------------------------------------
__global__ void gemm_ln_kernel(const float* __restrict__ A, int K,
                               const float* __restrict__ W,
                               const float* __restrict__ bias,
                               const float* __restrict__ resid,
                               const float* __restrict__ g,
                               const float* __restrict__ b,
                               float* __restrict__ out) {
  const int lane = threadIdx.x;
  const int m0 = blockIdx.x * 16;

  v8f acc[8];
#pragma unroll
  for (int c = 0; c < 8; ++c) acc[c] = vzero8();

  for (int k0 = 0; k0 < K; k0 += 4) {
    v2f a = load_a(A, K, m0, k0, lane);
#pragma unroll
    for (int c = 0; c < 8; ++c)
      acc[c] = wmma4(a, load_b(W, DM, k0, c * 16, lane), acc[c]);
  }

  const int rowoff = (lane >> 4) << 3;
  const int nlo = lane & 15;

#pragma unroll
  for (int c = 0; c < 8; ++c) {
    int col = c * 16 + nlo;
    float bb = bias ? bias[col] : 0.f;
#pragma unroll
    for (int r = 0; r < 8; ++r) {
      int row = m0 + r + rowoff;
      acc[c][r] += bb + resid[(size_t)row * DM + col];
    }
  }

  float mean[8], rstd[8];
#pragma unroll
  for (int r = 0; r < 8; ++r) {
    float s = 0.f;
#pragma unroll
    for (int c = 0; c < 8; ++c) s += acc[c][r];
    mean[r] = half_sum(s) * (1.f / DM);
  }
#pragma unroll
  for (int r = 0; r < 8; ++r) {
    float s = 0.f;
#pragma unroll
    for (int c = 0; c < 8; ++c) {
      float d = acc[c][r] - mean[r];
      s += d * d;
    }
    rstd[r] = rsqrtf(half_sum(s) * (1.f / DM) + LN_EPS);
  }
#pragma unroll
  for (int c = 0; c < 8; ++c) {
    int col = c * 16 + nlo;
    float gg = g[col], b2 = b[col];
#pragma unroll
    for (int r = 0; r < 8; ++r) {
      int row = m0 + r + rowoff;
      out[(size_t)row * DM + col] = (acc[c][r] - mean[r]) * rstd[r] * gg + b2;
    }
  }
}

// ---------------------------------------------------------------------------
// Kernel 4: FFN1, [4096,128]x[128,512] + bias + ReLU (LEAK==0).
// grid = (512/64, 4096/16), block = 32.
// ---------------------------------------------------------------------------
__global__ void ffn1_kernel(const float* __restrict__ X,
                            const float* __restrict__ W1,
                            const float* __restrict__ b1,
                            float* __restrict__ Hbuf) {
  const int lane = threadIdx.x;
  const int n0 = blockIdx.x * 64;
  const int m0 = blockIdx.y * 16;

  v8f acc[4];
#pragma unroll
  for (int c = 0; c < 4; ++c) acc[c] = vzero8();

  for (int k0 = 0; k0 < DM; k0 += 4) {
    v2f a = load_a(X, DM, m0, k0, lane);
#pragma unroll
    for (int c = 0; c < 4; ++c)
      acc[c] = wmma4(a, load_b(W1, DHID, k0, n0 + c * 16, lane), acc[c]);
  }

  const int rowoff = (lane >> 4) << 3;
  const int nlo = lane & 15;
#pragma unroll
  for (int c = 0; c < 4; ++c) {
    int col = n0 + c * 16 + nlo;
    float bb = b1[col];
#pragma unroll
    for (int r = 0; r < 8; ++r) {
      int row = m0 + r + rowoff;
      Hbuf[(size_t)row * DHID + col] = fmaxf(acc[c][r] + bb, 0.f);
    }
  }
}

// ---------------------------------------------------------------------------
// Host-side launcher.
// ---------------------------------------------------------------------------
extern "C" void kernel_launch(void* const* d_in, const int* in_sizes, int n_in,
                              void* d_out, int out_size, void* d_ws,
                              size_t ws_size, hipStream_t stream) {
  (void)in_sizes; (void)n_in; (void)out_size; (void)ws_size;

  const float* x   = (const float*)d_in[0];
  const float* Wq  = (const float*)d_in[1];
  const float* Wk  = (const float*)d_in[2];
  const float* Wv  = (const float*)d_in[3];
  const float* Wo  = (const float*)d_in[4];
  const float* g1  = (const float*)d_in[5];
  const float* be1 = (const float*)d_in[6];
  const float* W1  = (const float*)d_in[7];
  const float* b1  = (const float*)d_in[8];
  const float* W2  = (const float*)d_in[9];
  const float* b2  = (const float*)d_in[10];
  const float* g2  = (const float*)d_in[11];
  const float* be2 = (const float*)d_in[12];
  float* out = (float*)d_out;

  float* ws = (float*)d_ws;
  float* qkv  = ws;                                    // 3*H*N*64  (24 MB)
  float* Obuf = qkv + (size_t)3 * HEADS * NTOK * DQKV; // N*512     (8 MB)
  float* x1   = Obuf + (size_t)NTOK * QKVC;            // N*128     (2 MB)
  float* xA   = x1 + (size_t)NTOK * DM;                // N*128     (2 MB)
  float* Hbuf = qkv;  // FFN hidden reuses QKV region (free after attention)

  const float* xin = x;
  for (int l = 0; l < LAYERS; ++l) {
    const float* Wql = Wq + (size_t)l * DM * QKVC;
    const float* Wkl = Wk + (size_t)l * DM * QKVC;
    const float* Wvl = Wv + (size_t)l * DM * QKVC;
    const float* Wol = Wo + (size_t)l * QKVC * DM;
    const float* W1l = W1 + (size_t)l * DM * DHID;
    const float* b1l = b1 + (size_t)l * DHID;
    const float* W2l = W2 + (size_t)l * DHID * DM;
    const float* b2l = b2 + (size_t)l * DM;
    const float* g1l = g1 + (size_t)l * DM;
    const float* e1l = be1 + (size_t)l * DM;
    const float* g2l = g2 + (size_t)l * DM;
    const float* e2l = be2 + (size_t)l * DM;
    float* xout = (l == LAYERS - 1) ? out : xA;

    qkv_kernel<<<dim3(QKVC / 64, NTOK / 16, 3), 32, 0, stream>>>(
        xin, Wql, Wkl, Wvl, qkv);
    attn_kernel<<<dim3(NTOK / 16, HEADS), 32, 0, stream>>>(qkv, Obuf);
    gemm_ln_kernel<<<dim3(NTOK / 16), 32, 0, stream>>>(
        Obuf, QKVC, Wol, nullptr, xin, g1l, e1l, x1);
    ffn1_kernel<<<dim3(DHID / 64, NTOK / 16), 32, 0, stream>>>(
        x1, W1l, b1l, Hbuf);
    gemm_ln_kernel<<<dim3(NTOK / 16), 32, 0, stream>>>(
        Hbuf, DHID, W2l, b2l, x1, g2l, e2l, xout);
    xin = xout;
  }
}